// GCNPos_16381005267350
// MI455X (gfx1250) — compile-verified
//
#include <hip/hip_runtime.h>
#include <hip/hip_bf16.h>

typedef float v2f __attribute__((ext_vector_type(2)));
typedef float v8f __attribute__((ext_vector_type(8)));

#define NHID   64
#define CHUNK  512       // neighbor rows staged per block (137KB LDS -> 2 WGs/WGP)
#define ROWPAD 68        // LDS row stride (floats) -> conflict-free
#define PSTR   528       // per-block partial: 8 max + 8 dsum + 8*64 S
#define TILES  (CHUNK / 128)   // 16-row tiles per wave (8 waves)
#define NRED   64        // level-2 reduction blocks

// ---------------------------------------------------------------------------
// Prologue: conv(1,11,11)*(8,1,3,3) -> 648, pos@pos_w.T -> 16, relu(lin) -> h
// ---------------------------------------------------------------------------
__global__ __launch_bounds__(128) void prologue_kernel(
    const float* __restrict__ x, const float* __restrict__ pos,
    const float* __restrict__ conv_w, const float* __restrict__ pos_w,
    const float* __restrict__ lin_w, float* __restrict__ ws)
{
  __shared__ float y[648];
  __shared__ float pe[16];
  int t = threadIdx.x;
  for (int e = t; e < 648; e += 128) {
    int c = e / 81, rc = e % 81, r = rc / 9, col = rc % 9;
    float s = 0.f;
    for (int dr = 0; dr < 3; dr++)
      for (int dc = 0; dc < 3; dc++)
        s += x[(r + dr) * 11 + (col + dc)] * conv_w[c * 9 + dr * 3 + dc];
    y[e] = s;
  }
  if (t < 16) {
    float s = 0.f;
    for (int j = 0; j < 3; j++) s += pos[j] * pos_w[t * 3 + j];
    pe[t] = s;
  }
  __syncthreads();
  if (t < 64) {
    const float* L = lin_w + (size_t)t * 664;
    float s = 0.f;
    for (int e = 0; e < 648; e++) s += y[e] * L[e];
    for (int p = 0; p < 16; p++)  s += pe[p] * L[648 + p];
    ws[t] = fmaxf(s, 0.f);
  }
}

// ---------------------------------------------------------------------------
// Per relation: r[h][m] = (1/sqrt(8)) * sum_d q[h*8+d] * wk[h*8+d][m]
// q[j] = h . wq[j,:]     (wqkv[i] is (192,64): wq rows 0..63, wk 64..127)
// ---------------------------------------------------------------------------
__global__ __launch_bounds__(64) void relmat_kernel(
    float* __restrict__ ws, const float* __restrict__ wqkv, int rel)
{
  __shared__ float q[64];
  __shared__ float hs[64];
  int t = threadIdx.x;
  hs[t] = ws[t];
  __syncthreads();
  const float* wq = wqkv + (size_t)rel * 192 * 64;
  {
    float s = 0.f;
    const float* row = wq + (size_t)t * 64;
    for (int m = 0; m < 64; m++) s += hs[m] * row[m];
    q[t] = s;
  }
  __syncthreads();
  const float* wk = wq + 64 * 64;
  const float scale = 0.35355339059327373f;  // 1/sqrt(HEAD_DIM)
  float* R = ws + 64;
  int h = t >> 3, mb = (t & 7) * 8;
  for (int j = 0; j < 8; j++) {
    int m = mb + j;
    float s = 0.f;
    for (int d = 0; d < 8; d++) s += q[h * 8 + d] * wk[(h * 8 + d) * 64 + m];
    R[h * 64 + m] = s * scale;
  }
}

// ---------------------------------------------------------------------------
// Streaming attention pass. Each block stages CHUNK neighbor rows to LDS,
// computes scores with V_WMMA_F32_16X16X4_F32 (GEMM1: Nb(16x64) @ rT(64x16)),
// block-wide max, exp, and accumulates S = P @ Nb (GEMM2) also via WMMA.
// Emits per-block partials (max_h, sumexp_h, S_h[64]) for flash-style merge.
// 137KB LDS -> two workgroups per WGP: one stages from HBM while the other
// computes from LDS (double-buffering via co-residency).
// ---------------------------------------------------------------------------
__global__ __launch_bounds__(256) void attn_kernel(
    const float* __restrict__ nb, const float* __restrict__ Rm,
    float* __restrict__ part, int n)
{
  extern __shared__ float lds[];
  const int tid  = threadIdx.x;
  const int lane = tid & 31;
  const int wv   = tid >> 5;        // wave 0..7
  const int hi   = lane >> 4;       // lane half
  const int lh   = lane & 15;
  const int blockStart = blockIdx.x * CHUNK;

  // ---- stage CHUNK x 64 floats into LDS (zero-pad OOB rows) ----
  for (int i = tid; i < CHUNK * 16; i += 256) {
    int row = i >> 4, c4 = i & 15;
    int g = blockStart + row;
    float4 v = make_float4(0.f, 0.f, 0.f, 0.f);
    if (g < n) v = *(const float4*)(nb + (size_t)g * 64 + c4 * 4);
    float* d = &lds[row * ROWPAD + c4 * 4];
    d[0] = v.x; d[1] = v.y; d[2] = v.z; d[3] = v.w;
  }

  // ---- B operand for GEMM1: rT (64 x 16), heads >= 8 zero-padded ----
  v2f B1[16];
#pragma unroll
  for (int kk = 0; kk < 16; kk++) {
    int c = 4 * kk + 2 * hi;
    float bx = 0.f, by = 0.f;
    if (lh < 8) { bx = Rm[lh * 64 + c]; by = Rm[lh * 64 + c + 1]; }
    B1[kk].x = bx; B1[kk].y = by;
  }
  __syncthreads();

  float* wmaxs = &lds[CHUNK * ROWPAD];  // 64
  float* wdsum = wmaxs + 64;            // 64
  float* bmax  = wdsum + 64;            // 8
  float* bd    = bmax + 8;              // 8

  const int rowBase = wv * (16 * TILES);  // this wave's rows

  // ---- GEMM1: scores, keep C-tiles in registers; track per-lane max ----
  v8f st[TILES];
  float mlane = -1e30f;
#pragma unroll
  for (int t = 0; t < TILES; t++) {
    v8f c = {0.f, 0.f, 0.f, 0.f, 0.f, 0.f, 0.f, 0.f};
    const float* arow = &lds[(rowBase + t * 16 + lh) * ROWPAD];
#pragma unroll
    for (int kk = 0; kk < 16; kk++) {
      v2f a;
      a.x = arow[4 * kk + 2 * hi];
      a.y = arow[4 * kk + 2 * hi + 1];
      c = __builtin_amdgcn_wmma_f32_16x16x4_f32(
          false, a, false, B1[kk], (short)0, c, false, false);
    }
    st[t] = c;
#pragma unroll
    for (int r = 0; r < 8; r++) {
      int g = blockStart + rowBase + t * 16 + r + 8 * hi;
      float s = st[t][r];
      mlane = (g < n && s > mlane) ? s : mlane;
    }
  }

  // ---- block max per head ----
  float hm = fmaxf(mlane, __shfl_xor(mlane, 16));
  if (lane < 8) wmaxs[wv * 8 + lane] = hm;
  __syncthreads();
  if (tid < 8) {
    float m = wmaxs[tid];
    for (int w2 = 1; w2 < 8; w2++) m = fmaxf(m, wmaxs[w2 * 8 + tid]);
    bmax[tid] = m;
  }
  __syncthreads();
  float bm = bmax[lh & 7];

  // ---- exp + GEMM2: Sacc(16x64) += P(16x16) @ Nb(16x64) ----
  v8f Sacc[4];
#pragma unroll
  for (int tN = 0; tN < 4; tN++) Sacc[tN] = (v8f){0.f,0.f,0.f,0.f,0.f,0.f,0.f,0.f};
  float dsum = 0.f;

#pragma unroll
  for (int t = 0; t < TILES; t++) {
    v8f p;
#pragma unroll
    for (int r = 0; r < 8; r++) {
      int g = blockStart + rowBase + t * 16 + r + 8 * hi;
      bool valid = (g < n) && (lh < 8);
      float e = __expf(st[t][r] - bm);
      p[r] = valid ? e : 0.f;
      dsum += p[r];
    }
    const float* brow = &lds[(rowBase + t * 16) * ROWPAD];
#pragma unroll
    for (int kk = 0; kk < 4; kk++) {
      // re-layout P (C-layout) into A-operand layout: identity or xor-16 swap
      float xlo = p[(4 * kk) & 7],     xhi = p[(4 * kk + 2) & 7];
      float ylo = p[(4 * kk + 1) & 7], yhi = p[(4 * kk + 3) & 7];
      v2f a;
      if (kk < 2) {                       // source rows live in low lane half
        float t0 = __shfl_xor(xhi, 16);
        float t1 = __shfl_xor(yhi, 16);
        a.x = hi ? t0 : xlo;
        a.y = hi ? t1 : ylo;
      } else {                            // source rows live in high lane half
        float t0 = __shfl_xor(xlo, 16);
        float t1 = __shfl_xor(ylo, 16);
        a.x = hi ? xhi : t0;
        a.y = hi ? yhi : t1;
      }
      int rr = 4 * kk + 2 * hi;
#pragma unroll
      for (int tN = 0; tN < 4; tN++) {
        v2f b;
        int col = lh + 16 * tN;
        b.x = brow[rr * ROWPAD + col];
        b.y = brow[(rr + 1) * ROWPAD + col];
        Sacc[tN] = __builtin_amdgcn_wmma_f32_16x16x4_f32(
            false, a, false, b, (short)0, Sacc[tN], false, false);
      }
    }
  }

  // ---- per-head exp-sum ----
  float dtot = dsum + __shfl_xor(dsum, 16);
  if (lane < 8) wdsum[wv * 8 + lane] = dtot;
  __syncthreads();  // also: all waves done reading neighbor LDS
  if (tid < 8) {
    float s = 0.f;
    for (int w2 = 0; w2 < 8; w2++) s += wdsum[w2 * 8 + tid];
    bd[tid] = s;
  }

  // ---- reduce Sacc across waves (reuse neighbor LDS region) ----
  float* red = lds;  // 8 waves * 512 floats
  if (hi == 0) {     // rows 0..7 (= heads) live in lanes 0..15
#pragma unroll
    for (int tN = 0; tN < 4; tN++)
#pragma unroll
      for (int r = 0; r < 8; r++)
        red[wv * 512 + r * 64 + (lh + 16 * tN)] = Sacc[tN][r];
  }
  __syncthreads();

  float* bp = part + (size_t)blockIdx.x * PSTR;
  if (tid < 8) { bp[tid] = bmax[tid]; bp[8 + tid] = bd[tid]; }
  for (int i = tid; i < 512; i += 256) {
    float s = 0.f;
    for (int w2 = 0; w2 < 8; w2++) s += red[w2 * 512 + i];
    bp[16 + i] = s;
  }
}

// ---------------------------------------------------------------------------
// Parallel flash-merge of level-1 partials into NRED level-2 partials.
// Merge is associative: m' = max m_b ; d' = sum d_b exp(m_b - m') ;
// S' = sum S_b exp(m_b - m').
// ---------------------------------------------------------------------------
__global__ __launch_bounds__(512) void reduce_kernel(
    const float* __restrict__ pin, float* __restrict__ pout,
    int nblk, int per)
{
  __shared__ float M[8];
  int b0 = blockIdx.x * per;
  int b1 = b0 + per; if (b1 > nblk) b1 = nblk;
  int t = threadIdx.x;
  if (t < 8) {
    float m = -1e30f;
    for (int b = b0; b < b1; b++) m = fmaxf(m, pin[(size_t)b * PSTR + t]);
    float d = 0.f;
    for (int b = b0; b < b1; b++)
      d += pin[(size_t)b * PSTR + 8 + t] * __expf(pin[(size_t)b * PSTR + t] - m);
    M[t] = m;
    pout[(size_t)blockIdx.x * PSTR + t] = m;
    pout[(size_t)blockIdx.x * PSTR + 8 + t] = d;
  }
  __syncthreads();
  {
    int h = t >> 6;
    float mh = M[h], s = 0.f;
    for (int b = b0; b < b1; b++)
      s += pin[(size_t)b * PSTR + 16 + t] * __expf(pin[(size_t)b * PSTR + h] - mh);
    pout[(size_t)blockIdx.x * PSTR + 16 + t] = s;
  }
}

// ---------------------------------------------------------------------------
// Finalize: merge NRED level-2 partials, apply wv, wo, residual + LN,
// FFN + LN. Updates h in ws.
// ---------------------------------------------------------------------------
__global__ __launch_bounds__(512) void finalize_kernel(
    float* __restrict__ ws, const float* __restrict__ part,
    const float* __restrict__ wqkv, const float* __restrict__ wo,
    const float* __restrict__ g1, const float* __restrict__ b1,
    const float* __restrict__ g2, const float* __restrict__ b2,
    const float* __restrict__ f1, const float* __restrict__ f2,
    int rel, int nblk)
{
  __shared__ float M[8], D[8], S[512], o64[64], u[64], h1[64], ff[64], u2[64];
  float* H = ws;
  int t = threadIdx.x;

  if (t < 8) {
    float m = -1e30f;
    for (int b = 0; b < nblk; b++) m = fmaxf(m, part[(size_t)b * PSTR + t]);
    float d = 0.f;
    for (int b = 0; b < nblk; b++)
      d += part[(size_t)b * PSTR + 8 + t] * __expf(part[(size_t)b * PSTR + t] - m);
    M[t] = m; D[t] = d;
  }
  __syncthreads();
  {
    int h = t >> 6;
    float mh = M[h], s = 0.f;
    for (int b = 0; b < nblk; b++)
      s += part[(size_t)b * PSTR + 16 + t] * __expf(part[(size_t)b * PSTR + h] - mh);
    S[t] = s;
  }
  __syncthreads();
  if (t < 64) {  // o64[j] = (wv[j,:] . S[h,:]) / D[h],  j = h*8+d
    int h = t >> 3;
    const float* wvr = wqkv + ((size_t)rel * 192 + 128 + t) * 64;
    float s = 0.f;
    for (int m = 0; m < 64; m++) s += wvr[m] * S[h * 64 + m];
    o64[t] = s / D[h];
  }
  __syncthreads();
  if (t < 64) {  // o2 = o64 @ wo.T ; residual
    const float* wor = wo + ((size_t)rel * 64 + t) * 64;
    float s = 0.f;
    for (int k = 0; k < 64; k++) s += o64[k] * wor[k];
    u[t] = H[t] + s;
  }
  __syncthreads();
  if (t < 64) {  // LN1
    float mu = 0.f;
    for (int k = 0; k < 64; k++) mu += u[k];
    mu *= (1.f / 64.f);
    float var = 0.f;
    for (int k = 0; k < 64; k++) { float d = u[k] - mu; var += d * d; }
    var *= (1.f / 64.f);
    h1[t] = (u[t] - mu) * rsqrtf(var + 1e-5f) * g1[rel * 64 + t] + b1[rel * 64 + t];
  }
  __syncthreads();
  if (t < 64) {  // relu(h1 @ ff1.T)
    const float* r = f1 + ((size_t)rel * 64 + t) * 64;
    float s = 0.f;
    for (int k = 0; k < 64; k++) s += h1[k] * r[k];
    ff[t] = fmaxf(s, 0.f);
  }
  __syncthreads();
  if (t < 64) {  // @ ff2.T ; residual
    const float* r = f2 + ((size_t)rel * 64 + t) * 64;
    float s = 0.f;
    for (int k = 0; k < 64; k++) s += ff[k] * r[k];
    u2[t] = h1[t] + s;
  }
  __syncthreads();
  if (t < 64) {  // LN2 -> new h
    float mu = 0.f;
    for (int k = 0; k < 64; k++) mu += u2[k];
    mu *= (1.f / 64.f);
    float var = 0.f;
    for (int k = 0; k < 64; k++) { float d = u2[k] - mu; var += d * d; }
    var *= (1.f / 64.f);
    H[t] = (u2[t] - mu) * rsqrtf(var + 1e-5f) * g2[rel * 64 + t] + b2[rel * 64 + t];
  }
}

// ---------------------------------------------------------------------------
// Output: out[k] = h . out_w[k,:], k < 16384
// ---------------------------------------------------------------------------
__global__ __launch_bounds__(256) void out_kernel(
    const float* __restrict__ ws, const float* __restrict__ out_w,
    float* __restrict__ out)
{
  __shared__ float hs[64];
  if (threadIdx.x < 64) hs[threadIdx.x] = ws[threadIdx.x];
  __syncthreads();
  int k = blockIdx.x * 256 + threadIdx.x;
  const float* r = out_w + (size_t)k * 64;
  float s = 0.f;
#pragma unroll
  for (int m = 0; m < 64; m++) s += hs[m] * r[m];
  out[k] = s;
}

// ---------------------------------------------------------------------------
extern "C" void kernel_launch(void* const* d_in, const int* in_sizes, int n_in,
                              void* d_out, int out_size, void* d_ws, size_t ws_size,
                              hipStream_t stream) {
  const float* x      = (const float*)d_in[0];
  const float* pos    = (const float*)d_in[1];
  const float* nb     = (const float*)d_in[2];
  const float* conv_w = (const float*)d_in[3];
  const float* pos_w  = (const float*)d_in[4];
  const float* lin_w  = (const float*)d_in[5];
  const float* wqkv   = (const float*)d_in[6];
  const float* wo     = (const float*)d_in[7];
  const float* g1     = (const float*)d_in[8];
  const float* b1     = (const float*)d_in[9];
  const float* g2     = (const float*)d_in[10];
  const float* b2     = (const float*)d_in[11];
  const float* f1     = (const float*)d_in[12];
  const float* f2     = (const float*)d_in[13];
  const float* out_w  = (const float*)d_in[14];
  float* ws = (float*)d_ws;

  int n = in_sizes[2] / NHID;                 // 1,000,000 neighbors
  int nblk = (n + CHUNK - 1) / CHUNK;
  int per  = (nblk + NRED - 1) / NRED;
  float* part1 = ws + 1024;                         // level-1 partials
  float* part2 = part1 + (size_t)nblk * PSTR;       // level-2 partials
  size_t lds_bytes = (size_t)(CHUNK * ROWPAD + 160) * sizeof(float);
  (void)hipFuncSetAttribute((const void*)attn_kernel,
                            hipFuncAttributeMaxDynamicSharedMemorySize,
                            (int)lds_bytes);

  prologue_kernel<<<1, 128, 0, stream>>>(x, pos, conv_w, pos_w, lin_w, ws);
  for (int i = 0; i < 16; i++) {
    relmat_kernel<<<1, 64, 0, stream>>>(ws, wqkv, i);
    attn_kernel<<<nblk, 256, lds_bytes, stream>>>(nb, ws + 64, part1, n);
    reduce_kernel<<<NRED, 512, 0, stream>>>(part1, part2, nblk, per);
    finalize_kernel<<<1, 512, 0, stream>>>(ws, part2, wqkv, wo,
                                           g1, b1, g2, b2, f1, f2, i, NRED);
  }
  out_kernel<<<64, 256, 0, stream>>>(ws, out_w, (float*)d_out);
}